// SSMLayer_73624329388542
// MI455X (gfx1250) — compile-verified
//
#include <hip/hip_runtime.h>
#include <stdint.h>

#define MDIM 2048
#define DDIM 64
#define TILE 64
#define TPITCH 68   // transpose-staging row pitch: 68*4 B = 272 B (16B aligned, 68 % 64 = 4 -> conflict-free transposed reads)

// ---- CDNA5 async global->LDS helpers (gfx1250) ------------------------------
typedef __attribute__((address_space(3))) void lds_void_t;

__device__ __forceinline__ unsigned lds_byte_offset(void* p) {
    // generic -> LDS addrspace cast strips the aperture, leaving the 32-bit
    // offset within the workgroup's LDS allocation (what VDST of the async
    // load expects as its LDS destination address).
    return (unsigned)(unsigned long long)(lds_void_t*)p;
}

__device__ __forceinline__ void async_copy_b128(void* lds_p, const void* gptr) {
    unsigned loff = lds_byte_offset(lds_p);
    // GLOBAL_LOAD_ASYNC_TO_LDS_B128 (opcode 98): 16 bytes/lane, tracked by ASYNCcnt.
    asm volatile("global_load_async_to_lds_b128 %0, %1, off"
                 :
                 : "v"(loff), "v"(gptr)
                 : "memory");
}

__device__ __forceinline__ void wait_async_zero() {
    asm volatile("s_wait_asynccnt 0" ::: "memory");
}

// ---- kernel 1: init global min/max cells in workspace -----------------------
__global__ void ssm_init_minmax(unsigned* __restrict__ mmx) {
    mmx[0] = 0x7F800000u; // +inf bits  (running min; uint order == float order for x>=0)
    mmx[1] = 0x00000000u; // 0.0f bits  (running max)
}

// ---- kernel 2: pairwise L1 distance (upper-triangle tiles) + min/max --------
__global__ __launch_bounds__(256) void ssm_dist_kernel(const float* __restrict__ x,
                                                       float* __restrict__ dist,
                                                       unsigned* __restrict__ mmx) {
    const int bi = blockIdx.y;   // row-tile
    const int bj = blockIdx.x;   // col-tile
    if (bi > bj) return;         // symmetry: only upper-triangle tiles (uniform exit)

    // One shared pool: panels sA|sB (2*16KB), later reused for the 64x64
    // transpose staging (needs 64*68 floats = 17 KB) and the min/max reduction.
    __shared__ __align__(16) float smem[2 * TILE * TILE];
    float* sA = smem;               // [d][i] panel
    float* sB = smem + TILE * TILE; // [d][j] panel

    const int i0  = bi * TILE;
    const int j0  = bj * TILE;
    const int tid = threadIdx.x;

    // Stage both 64x64 fp32 panels via the async tensor-data path.
    // 1024 float4 per panel / 256 threads -> 4 rounds, 16B per lane per round.
#pragma unroll
    for (int r = 0; r < 4; ++r) {
        const int q  = r * 256 + tid;   // [0, 1024)
        const int d  = q >> 4;          // row (dimension)
        const int c4 = (q & 15) << 2;   // col within tile, multiple of 4
        async_copy_b128(&sA[d * TILE + c4], &x[(size_t)d * MDIM + i0 + c4]);
        async_copy_b128(&sB[d * TILE + c4], &x[(size_t)d * MDIM + j0 + c4]);
    }
    wait_async_zero();   // this wave's async copies landed in LDS
    __syncthreads();     // all waves' copies visible

    const int tj = tid & 15;  // 16 thread-cols
    const int ti = tid >> 4;  // 16 thread-rows

    float acc[4][4];
#pragma unroll
    for (int r = 0; r < 4; ++r)
#pragma unroll
        for (int c = 0; c < 4; ++c) acc[r][c] = 0.0f;

#pragma unroll 4
    for (int d = 0; d < DDIM; ++d) {
        const float4 a = *(const float4*)&sA[d * TILE + ti * 4];
        const float4 b = *(const float4*)&sB[d * TILE + tj * 4];
        const float ar[4] = {a.x, a.y, a.z, a.w};
        const float br[4] = {b.x, b.y, b.z, b.w};
#pragma unroll
        for (int r = 0; r < 4; ++r)
#pragma unroll
            for (int c = 0; c < 4; ++c)
                acc[r][c] += __builtin_fabsf(ar[r] - br[c]); // v_sub + v_add |..|
    }

    // Direct tile: coalesced b128 rows, plus local min/max fold.
    float lmin = __uint_as_float(0x7F800000u);
    float lmax = 0.0f;
#pragma unroll
    for (int r = 0; r < 4; ++r) {
        float4 v;
        v.x = acc[r][0]; v.y = acc[r][1]; v.z = acc[r][2]; v.w = acc[r][3];
        *(float4*)&dist[(size_t)(i0 + ti * 4 + r) * MDIM + (j0 + tj * 4)] = v;
        lmin = fminf(lmin, fminf(fminf(v.x, v.y), fminf(v.z, v.w)));
        lmax = fmaxf(lmax, fmaxf(fmaxf(v.x, v.y), fmaxf(v.z, v.w)));
    }

    // Mirrored tile (bi < bj): transpose through LDS so global writes stay coalesced.
    if (bi != bj) {             // uniform branch per block
        __syncthreads();        // everyone done reading panels
#pragma unroll
        for (int r = 0; r < 4; ++r) {
            float4 v;
            v.x = acc[r][0]; v.y = acc[r][1]; v.z = acc[r][2]; v.w = acc[r][3];
            *(float4*)&smem[(ti * 4 + r) * TPITCH + tj * 4] = v;  // 16B-aligned b128
        }
        __syncthreads();
#pragma unroll
        for (int r = 0; r < 4; ++r) {
            float4 v;  // tile[q][p] with p = ti*4+r (new row, j-dim), q = tj*4+c (new col, i-dim)
            v.x = smem[(tj * 4 + 0) * TPITCH + ti * 4 + r];
            v.y = smem[(tj * 4 + 1) * TPITCH + ti * 4 + r];
            v.z = smem[(tj * 4 + 2) * TPITCH + ti * 4 + r];
            v.w = smem[(tj * 4 + 3) * TPITCH + ti * 4 + r];
            *(float4*)&dist[(size_t)(j0 + ti * 4 + r) * MDIM + (i0 + tj * 4)] = v;
        }
    }

    // Block reduction of min/max (reuse smem; symmetry => computed half covers all values).
    __syncthreads();
    smem[tid]       = lmin;
    smem[256 + tid] = lmax;
    __syncthreads();
#pragma unroll
    for (int s = 128; s > 0; s >>= 1) {
        if (tid < s) {
            smem[tid]       = fminf(smem[tid], smem[tid + s]);
            smem[256 + tid] = fmaxf(smem[256 + tid], smem[256 + tid + s]);
        }
        __syncthreads();
    }
    if (tid == 0) {
        // distances are >= 0 finite, so uint compare == float compare
        atomicMin(&mmx[0], __float_as_uint(smem[0]));
        atomicMax(&mmx[1], __float_as_uint(smem[256]));
    }
}

// ---- kernel 3: similarity = (mx - d) / (mx - mn) ----------------------------
__global__ __launch_bounds__(256) void ssm_norm_kernel(float* __restrict__ out,
                                                       const unsigned* __restrict__ mmx,
                                                       int n4) {
    const float mn  = __uint_as_float(mmx[0]);
    const float mx  = __uint_as_float(mmx[1]);
    const float inv = 1.0f / (mx - mn);

    float4* o4 = (float4*)out;
    int idx    = blockIdx.x * blockDim.x + threadIdx.x;
    int stride = gridDim.x * blockDim.x;
    for (int i = idx; i < n4; i += stride) {
        float4 v = o4[i];
        v.x = (mx - v.x) * inv;
        v.y = (mx - v.y) * inv;
        v.z = (mx - v.z) * inv;
        v.w = (mx - v.w) * inv;
        o4[i] = v;
    }
}

// ---- host-side launcher -----------------------------------------------------
extern "C" void kernel_launch(void* const* d_in, const int* in_sizes, int n_in,
                              void* d_out, int out_size, void* d_ws, size_t ws_size,
                              hipStream_t stream) {
    (void)in_sizes; (void)n_in; (void)out_size; (void)ws_size;

    const float* x   = (const float*)d_in[0];
    float*       out = (float*)d_out;
    unsigned*    mmx = (unsigned*)d_ws;

    ssm_init_minmax<<<dim3(1), dim3(1), 0, stream>>>(mmx);

    dim3 grid(MDIM / TILE, MDIM / TILE); // 32 x 32 tiles; lower triangle exits early
    ssm_dist_kernel<<<grid, dim3(256), 0, stream>>>(x, out, mmx);

    const int n4 = (MDIM * MDIM) / 4;
    ssm_norm_kernel<<<dim3(1024), dim3(256), 0, stream>>>(out, mmx, n4);
}